// GATLayer_39367670235251
// MI455X (gfx1250) — compile-verified
//
#include <hip/hip_runtime.h>

typedef __attribute__((ext_vector_type(2))) float v2f;
typedef __attribute__((ext_vector_type(8))) float v8f;
typedef __attribute__((ext_vector_type(4))) int   v4i;

#define B_       2
#define N_       2048
#define FIN_     128
#define H_       8
#define F_       32
#define C_       256          // H_*F_
#define NEGINF_  (-1.0e9f)
#define HT_PITCH 264          // 16B-aligned rows; stride mod 64 banks = 8 (conflict-free halves)

#define AS1 __attribute__((address_space(1)))
#define AS3 __attribute__((address_space(3)))

#ifndef __has_builtin
#define __has_builtin(x) 0
#endif
#if __has_builtin(__builtin_amdgcn_global_load_async_to_lds_b128)
#define HAVE_ASYNC_B128 1
#else
#define HAVE_ASYNC_B128 0
#endif
#if __has_builtin(__builtin_amdgcn_global_load_async_to_lds_b32)
#define HAVE_ASYNC_B32 1
#else
#define HAVE_ASYNC_B32 0
#endif

__device__ __forceinline__ float lrelu(float x) { return x > 0.f ? x : 0.2f * x; }

__device__ __forceinline__ v8f wmma_f32(v2f a, v2f b, v8f c) {
  // D = A(16x4 f32) * B(4x16 f32) + C(16x16 f32)
  return __builtin_amdgcn_wmma_f32_16x16x4_f32(false, a, false, b, (short)0, c,
                                               false, false);
}

__device__ __forceinline__ void wait_async() {
#if HAVE_ASYNC_B128 || HAVE_ASYNC_B32
#if __has_builtin(__builtin_amdgcn_s_wait_asynccnt)
  __builtin_amdgcn_s_wait_asynccnt(0);
#else
  asm volatile("s_wait_asynccnt 0x0" ::: "memory");
#endif
#endif
}

// ---------------------------------------------------------------------------
// Kernel 1: ht = h @ W   (4096x128 @ 128x256), one wave per 16x16 output tile
// ---------------------------------------------------------------------------
__global__ void __launch_bounds__(128) gat_proj(const float* __restrict__ h,
                                                const float* __restrict__ W,
                                                float* __restrict__ ht) {
  const int wave = threadIdx.x >> 5;
  const int lane = threadIdx.x & 31;
  const int half = lane >> 4;
  const int lr   = lane & 15;

  const int tile = blockIdx.x * 4 + wave;   // 4096 tiles total
  const int tm   = tile >> 4;               // 256 row tiles (B*N/16)
  const int tn   = tile & 15;               // 16 col tiles (C/16)
  const int row  = tm * 16 + lr;
  const int col  = tn * 16 + lr;

  v8f acc = {};
#pragma unroll 4
  for (int k0 = 0; k0 < FIN_; k0 += 4) {
    const int ka = k0 + 2 * half;
    v2f a = *(const v2f*)(h + row * FIN_ + ka);   // A: rows striped in lane
    v2f b;
    b.x = W[(ka + 0) * C_ + col];                 // B: cols striped across lanes
    b.y = W[(ka + 1) * C_ + col];
    acc = wmma_f32(a, b, acc);
  }
#pragma unroll
  for (int v = 0; v < 8; ++v) {
    const int r = tm * 16 + v + half * 8;         // C/D layout: vgpr v -> row v / v+8
    ht[r * C_ + col] = acc[v];
  }
}

// ---------------------------------------------------------------------------
// Kernel 2: s_i[b,h,n] = ht[b,n,h,:].a_i ; s_j likewise. One wave per (b,n).
// ---------------------------------------------------------------------------
__global__ void __launch_bounds__(256) gat_scores(const float* __restrict__ ht,
                                                  const float* __restrict__ attn,
                                                  float* __restrict__ si,
                                                  float* __restrict__ sj) {
  const int wave = threadIdx.x >> 5;
  const int lane = threadIdx.x & 31;
  const int bn   = blockIdx.x * 8 + wave;     // 0 .. B*N-1
  const int b    = bn >> 11;
  const int n    = bn & (N_ - 1);

  const float ai = attn[lane];                // a_i = attn[0..31]
  const float aj = attn[F_ + lane];           // a_j = attn[32..63]

#pragma unroll
  for (int hh = 0; hh < H_; ++hh) {
    const float v = ht[bn * C_ + hh * F_ + lane];
    float vi = v * ai;
    float vj = v * aj;
#pragma unroll
    for (int off = 16; off > 0; off >>= 1) {
      vi += __shfl_xor(vi, off, 32);
      vj += __shfl_xor(vj, off, 32);
    }
    if (lane == 0) {
      si[(b * H_ + hh) * N_ + n] = vi;
      sj[(b * H_ + hh) * N_ + n] = vj;
    }
  }
}

// ---------------------------------------------------------------------------
// Cooperative stage of one j-tile into LDS: 16x256 floats of ht + 16x16 adj.
// 256 threads: thread t covers ht row (t>>4), cols (t&15)*16..+15, and one adj.
// ---------------------------------------------------------------------------
__device__ __forceinline__ void stage_tile(const float* __restrict__ ght,
                                           const int* __restrict__ gadj,
                                           float* lht, int* ladj, int t) {
  const int jr = t >> 4;
  const int cs = (t & 15) * 16;
  const float* g = ght + jr * C_ + cs;
  float*       l = lht + jr * HT_PITCH + cs;
#if HAVE_ASYNC_B128
#pragma unroll
  for (int k = 0; k < 4; ++k)
    __builtin_amdgcn_global_load_async_to_lds_b128(
        (AS1 v4i*)(uintptr_t)(g + 4 * k),
        (AS3 v4i*)(uint32_t)(uintptr_t)(l + 4 * k), 0, 0);
#else
#pragma unroll
  for (int k = 0; k < 4; ++k)
    *(float4*)(l + 4 * k) = *(const float4*)(g + 4 * k);
#endif
  // adjacency tile: row (t>>4), col (t&15) -> ladj[t]
#if HAVE_ASYNC_B32
  __builtin_amdgcn_global_load_async_to_lds_b32(
      (AS1 int*)(uintptr_t)(gadj + jr * N_ + (t & 15)),
      (AS3 int*)(uint32_t)(uintptr_t)(ladj + t), 0, 0);
#else
  ladj[t] = gadj[jr * N_ + (t & 15)];
#endif
}

// ---------------------------------------------------------------------------
// Kernel 3: fused masked softmax + aggregation.
// One block = one (b, 16-row i-tile); its 8 waves = the 8 heads.
//   Pass A: block-cooperative masked row-max of s_j for all heads (adj read 1x).
//   Pass B: double-buffered LDS staging of ht/adj j-tiles (async-to-LDS when
//           available), each head-wave runs 8 v_wmma_f32_16x16x4 per tile.
// ---------------------------------------------------------------------------
__global__ void __launch_bounds__(256) gat_attn(const float* __restrict__ ht,
                                                const int* __restrict__ adj,
                                                const float* __restrict__ si,
                                                const float* __restrict__ sj,
                                                float* __restrict__ out) {
  __shared__ __align__(16) float lds_ht[2][16 * HT_PITCH];
  __shared__ __align__(16) int   lds_adj[2][16 * 16];
  __shared__ float lds_max[16][H_];

  const int t    = threadIdx.x;
  const int lane = t & 31;
  const int half = lane >> 4;
  const int lr   = lane & 15;
  const int hh   = t >> 5;            // wave id == head id

  const int b  = blockIdx.x >> 7;     // 256 blocks = B * (N/16)
  const int i0 = (blockIdx.x & 127) * 16;

  // ---- Pass A: masked max of s_j per (row, head); adj scanned once per block
  {
    const int rA = t >> 4;            // row 0..15 (16 threads each)
    const int cA = t & 15;            // j phase
    const int* arow = adj + (b * N_ + i0 + rA) * N_;
    float mx[H_];
#pragma unroll
    for (int k = 0; k < H_; ++k) mx[k] = -3.0e38f;
    for (int j = cA; j < N_; j += 16) {
      if (arow[j]) {
#pragma unroll
        for (int k = 0; k < H_; ++k) mx[k] = fmaxf(mx[k], sj[(b * H_ + k) * N_ + j]);
      }
    }
    // reduce over the 16 j-phases (lanes differing in low 4 bits share a row)
#pragma unroll
    for (int off = 1; off < 16; off <<= 1) {
#pragma unroll
      for (int k = 0; k < H_; ++k) mx[k] = fmaxf(mx[k], __shfl_xor(mx[k], off, 32));
    }
    if (cA == 0) {
#pragma unroll
      for (int k = 0; k < H_; ++k) lds_max[rA][k] = mx[k];
    }
  }
  __syncthreads();

  const float* sjrow = sj + (b * H_ + hh) * N_;
  const float  si_r  = si[(b * H_ + hh) * N_ + i0 + lr];
  const float  mraw  = lds_max[lr][hh];
  const float  mrow  = (mraw > -1.0e37f) ? lrelu(si_r + mraw) : NEGINF_;

  // ---- Pass B: double-buffered tile loop
  const int*   adjb = adj + (b * N_ + i0) * N_;
  const float* htb  = ht + (size_t)b * N_ * C_;
  const int    c0   = hh * F_ + lr;

  v8f acc0 = {}, acc1 = {};
  float lsum = 0.f;

  stage_tile(htb, adjb, lds_ht[0], lds_adj[0], t);
  wait_async();
  __syncthreads();

  for (int jt = 0; jt < N_ / 16; ++jt) {
    const int p = jt & 1;
    if (jt + 1 < N_ / 16)
      stage_tile(htb + (jt + 1) * 16 * C_, adjb + (jt + 1) * 16,
                 lds_ht[p ^ 1], lds_adj[p ^ 1], t);

    const int    j0  = jt * 16;
    const float* lht = lds_ht[p];
    const int*   lad = lds_adj[p];

    v2f ap[4];
#pragma unroll
    for (int kk = 0; kk < 4; ++kk) {
      const int jj = 4 * kk + 2 * half;
      const float e0 = lad[lr * 16 + jj]     ? lrelu(si_r + sjrow[j0 + jj])     : NEGINF_;
      const float e1 = lad[lr * 16 + jj + 1] ? lrelu(si_r + sjrow[j0 + jj + 1]) : NEGINF_;
      const float p0 = __expf(e0 - mrow);   // all-masked row: m = NEGINF -> p = 1
      const float p1 = __expf(e1 - mrow);
      ap[kk].x = p0;
      ap[kk].y = p1;
      lsum += p0 + p1;
    }
#pragma unroll
    for (int kk = 0; kk < 4; ++kk) {
      const int jj = 4 * kk + 2 * half;
      v2f b0, b1;
      b0.x = lht[jj * HT_PITCH + c0];
      b0.y = lht[(jj + 1) * HT_PITCH + c0];
      b1.x = lht[jj * HT_PITCH + c0 + 16];
      b1.y = lht[(jj + 1) * HT_PITCH + c0 + 16];
      acc0 = wmma_f32(ap[kk], b0, acc0);
      acc1 = wmma_f32(ap[kk], b1, acc1);
    }
    wait_async();
    __syncthreads();
  }

  // ---- Row sum: lanes L and L^16 hold the same row, disjoint j subsets
  lsum += __shfl_xor(lsum, 16, 32);
  const float inv = 1.0f / lsum;

#pragma unroll
  for (int v = 0; v < 8; ++v) {
    const int   srcr  = v + half * 8;
    const float scale = __shfl(inv, srcr, 32);
    float* op = out + (b * N_ + i0 + srcr) * C_ + hh * F_ + lr;
    op[0]  = acc0[v] * scale;
    op[16] = acc1[v] * scale;
  }
}

// ---------------------------------------------------------------------------
extern "C" void kernel_launch(void* const* d_in, const int* in_sizes, int n_in,
                              void* d_out, int out_size, void* d_ws, size_t ws_size,
                              hipStream_t stream) {
  const float* h    = (const float*)d_in[0];  // (B,N,Fin)
  const int*   adj  = (const int*)d_in[1];    // (B,N,N)
  const float* W    = (const float*)d_in[2];  // (Fin, H*F)
  const float* attn = (const float*)d_in[3];  // (2F, 1)
  float*       out  = (float*)d_out;          // (B,N,H*F)

  float* ht = (float*)d_ws;                         // B*N*C floats (4 MB)
  float* si = ht + (size_t)B_ * N_ * C_;            // B*H*N floats
  float* sj = si + (size_t)B_ * H_ * N_;            // B*H*N floats

  gat_proj<<<1024, 128, 0, stream>>>(h, W, ht);
  gat_scores<<<512, 256, 0, stream>>>(ht, attn, si, sj);
  gat_attn<<<256, 256, 0, stream>>>(ht, adj, si, sj, out);
}